// MulticlassAUCLoss_70815420776456
// MI455X (gfx1250) — compile-verified
//
#include <hip/hip_runtime.h>
#include <hip/hip_bf16.h>

typedef __attribute__((ext_vector_type(2))) float v2f;
typedef __attribute__((ext_vector_type(8))) float v8f;

#define AUC_ALPHA 20.0f
#define BIGV      1e30f
#define CCOUNT    8

// ---------------------------------------------------------------------------
// Kernel 0: zero the per-class accumulators (must run every launch; harness
// does not re-zero the workspace between replays).
// ---------------------------------------------------------------------------
__global__ void auc_init(double* __restrict__ sums, int* __restrict__ npos) {
  int t = threadIdx.x;
  if (t < CCOUNT) {
    sums[t] = 0.0;
    npos[t] = 0;
  }
}

// ---------------------------------------------------------------------------
// Kernel 1: per-sample softmax; emit masked exponentials
//   a[c][i] = (target==c) ? exp(-alpha*p) : BIG    (row side, positives)
//   b[c][i] = (target==c) ? BIG           : exp(+alpha*p)   (col side, negatives)
// so that 1/(1 + a_i*b_j) == pos_i*neg_j*sigmoid(alpha*(p_i-p_j)) up to ~1e-21.
// Also count positives per class.
// ---------------------------------------------------------------------------
__global__ void auc_prep(const float* __restrict__ logits,
                         const int*  __restrict__ target,
                         float* __restrict__ Aarr,
                         float* __restrict__ Barr,
                         int*   __restrict__ npos,
                         int n) {
  int i = blockIdx.x * blockDim.x + threadIdx.x;
  if (i >= n) return;

  float x[CCOUNT];
  float m = -3.4e38f;
#pragma unroll
  for (int c = 0; c < CCOUNT; ++c) {
    x[c] = logits[(size_t)i * CCOUNT + c];
    m = fmaxf(m, x[c]);
  }
  float s = 0.0f;
#pragma unroll
  for (int c = 0; c < CCOUNT; ++c) {
    x[c] = expf(x[c] - m);
    s += x[c];
  }
  float inv = 1.0f / s;
  int t = target[i];
#pragma unroll
  for (int c = 0; c < CCOUNT; ++c) {
    float p   = x[c] * inv;
    bool  pos = (t == c);
    Aarr[(size_t)c * n + i] = pos ? expf(-AUC_ALPHA * p) : BIGV;
    Barr[(size_t)c * n + i] = pos ? BIGV : expf(AUC_ALPHA * p);
  }
  atomicAdd(&npos[t], 1);
}

// Masked load: clamp address, substitute BIG for OOB (masked terms -> ~0).
__device__ __forceinline__ float load_masked(const float* __restrict__ p,
                                             int idx, int n) {
  int   idxc = (idx < n) ? idx : (n - 1);
  float v    = p[idxc];
  return (idx < n) ? v : BIGV;
}

// ---------------------------------------------------------------------------
// Kernel 2: pairwise tile sweep.
// Block = 256 threads = 8 waves. blockIdx.x = row tile, blockIdx.y = class.
// Each wave builds the A operand once (a in the K=0 column of the 16x4 A
// matrix), then sweeps its 64 column tiles two at a time: one 32-lane-wide
// b-load covers two 16-wide tiles (lanes 16-31 shuffled down by 16 for the
// second tile -> the load feeds all lanes and cannot be predicated), then
// two v_wmma_f32_16x16x4_f32 with C = ones give D[m][n] = 1 + a_m*b_n, and
// 16 v_rcp feed two independent packed-add accumulator banks.
// ---------------------------------------------------------------------------
__global__ void auc_pair(const float* __restrict__ Aarr,
                         const float* __restrict__ Barr,
                         double* __restrict__ sums,
                         int n, int tilesPerWave) {
  const int lane    = threadIdx.x & 31;
  const int wave    = threadIdx.x >> 5;
  const int rowTile = blockIdx.x;
  const int cls     = blockIdx.y;

  const float* __restrict__ a = Aarr + (size_t)cls * n;
  const float* __restrict__ b = Barr + (size_t)cls * n;

  const bool loLane = (lane < 16);

  // A operand (16x4 f32, 2 VGPRs): K=0 column = a-row values, rest 0.
  int   row  = rowTile * 16 + (lane & 15);
  float araw = load_masked(a, row, n);
  float av   = loLane ? araw : 0.0f;
  v2f Aop;
  Aop[0] = av;
  Aop[1] = 0.0f;

  v8f Cone;
#pragma unroll
  for (int r = 0; r < 8; ++r) Cone[r] = 1.0f;

  // Two independent accumulator banks (16 chains total).
  v8f acc0, acc1;
#pragma unroll
  for (int r = 0; r < 8; ++r) { acc0[r] = 0.0f; acc1[r] = 0.0f; }

  const int colBase = wave * tilesPerWave * 16;
  const int pairs   = tilesPerWave >> 1;

  for (int p = 0; p < pairs; ++p) {
    // One wave-wide load covers two 16-column tiles.
    int   col  = colBase + p * 32 + lane;
    float bval = load_masked(b, col, n);       // all 32 lanes: distinct data
    float bhi  = __shfl_down(bval, 16, 32);    // tile B's 16 values -> lanes 0-15

    v2f BopA, BopB;
    BopA[0] = loLane ? bval : 0.0f;  BopA[1] = 0.0f;
    BopB[0] = loLane ? bhi  : 0.0f;  BopB[1] = 0.0f;

    v8f D0 = __builtin_amdgcn_wmma_f32_16x16x4_f32(
        false, Aop, false, BopA, (short)0, Cone, false, false);
    v8f D1 = __builtin_amdgcn_wmma_f32_16x16x4_f32(
        false, Aop, false, BopB, (short)0, Cone, false, false);

#pragma unroll
    for (int r = 0; r < 8; ++r) acc0[r] += __builtin_amdgcn_rcpf(D0[r]);
#pragma unroll
    for (int r = 0; r < 8; ++r) acc1[r] += __builtin_amdgcn_rcpf(D1[r]);
  }

  // Odd-tile tail (not taken for n = 8192).
  if (tilesPerWave & 1) {
    int   col  = colBase + pairs * 32 + (lane & 15);
    float bval = load_masked(b, col, n);
    v2f Bop;
    Bop[0] = loLane ? bval : 0.0f;
    Bop[1] = 0.0f;
    v8f D = __builtin_amdgcn_wmma_f32_16x16x4_f32(
        false, Aop, false, Bop, (short)0, Cone, false, false);
#pragma unroll
    for (int r = 0; r < 8; ++r) acc0[r] += __builtin_amdgcn_rcpf(D[r]);
  }

  // Collapse the 16 accumulators (pairwise tree keeps chains short).
#pragma unroll
  for (int r = 0; r < 8; ++r) acc0[r] += acc1[r];
  float s01 = acc0[0] + acc0[1];
  float s23 = acc0[2] + acc0[3];
  float s45 = acc0[4] + acc0[5];
  float s67 = acc0[6] + acc0[7];
  float acc = (s01 + s23) + (s45 + s67);

  // wave32 reduction
#pragma unroll
  for (int off = 16; off > 0; off >>= 1)
    acc += __shfl_down(acc, off, 32);

  __shared__ float wsum[8];
  if (lane == 0) wsum[wave] = acc;
  __syncthreads();

  if (threadIdx.x == 0) {
    float ssum = 0.0f;
#pragma unroll
    for (int w = 0; w < 8; ++w) ssum += wsum[w];
    atomicAdd(&sums[cls], (double)ssum);   // global_atomic_add_f64
  }
}

// ---------------------------------------------------------------------------
// Kernel 3: assemble final scalar loss.
// ---------------------------------------------------------------------------
__global__ void auc_final(const double* __restrict__ sums,
                          const int*    __restrict__ npos,
                          float* __restrict__ out, int n) {
  if (threadIdx.x != 0 || blockIdx.x != 0) return;
  double total = 0.0;
  int    valid = 0;
  for (int c = 0; c < CCOUNT; ++c) {
    double np = (double)npos[c];
    double nn = (double)(n - npos[c]);
    if (np > 0.0 && nn > 0.0) {
      double denom = np * nn;
      if (denom < 1.0) denom = 1.0;
      total += sums[c] / denom;
      valid++;
    }
  }
  out[0] = (valid > 0) ? (float)(1.0 - total / (double)valid) : 0.0f;
}

// ---------------------------------------------------------------------------
// Host entry
// ---------------------------------------------------------------------------
extern "C" void kernel_launch(void* const* d_in, const int* in_sizes, int n_in,
                              void* d_out, int out_size, void* d_ws, size_t ws_size,
                              hipStream_t stream) {
  (void)n_in; (void)out_size; (void)ws_size;

  const float* logits = (const float*)d_in[0];
  const int*   target = (const int*)d_in[1];
  float*       out    = (float*)d_out;

  const int n = in_sizes[0] / CCOUNT;   // 8192

  char*   ws      = (char*)d_ws;
  size_t  abBytes = (size_t)CCOUNT * (size_t)n * sizeof(float);  // 256 KB each
  float*  Aarr    = (float*)(ws);
  float*  Barr    = (float*)(ws + abBytes);
  double* sums    = (double*)(ws + 2 * abBytes);
  int*    npos    = (int*)(ws + 2 * abBytes + CCOUNT * sizeof(double));

  auc_init<<<1, 32, 0, stream>>>(sums, npos);
  auc_prep<<<(n + 255) / 256, 256, 0, stream>>>(logits, target, Aarr, Barr, npos, n);

  const int rowTiles     = (n + 15) / 16;            // 512
  const int colTiles     = (n + 15) / 16;            // 512
  const int tilesPerWave = (colTiles + 7) / 8;       // 64 (OOB cols masked to ~0)

  dim3 grid(rowTiles, CCOUNT);
  auc_pair<<<grid, 256, 0, stream>>>(Aarr, Barr, sums, n, tilesPerWave);

  auc_final<<<1, 1, 0, stream>>>(sums, npos, out, n);
}